// LightGCN_36326833389965
// MI455X (gfx1250) — compile-verified
//
#include <hip/hip_runtime.h>
#include <hip/hip_bf16.h>

// ---------------- problem constants ----------------
#define NUSERS 8000
#define NITEMS 4000
#define NN     12000            // total nodes
#define DIM    64               // embedding dim
#define NLAYERS 3
#define NEDGES 300000
#define WPR    375              // meaningful 32-bit words per bitmap row (12000/32)
#define WPRP   376              // padded words per row -> 1504 B, 16-B aligned rows
#define ROWB   1504             // bytes per bitmap row
#define NB     750              // 16-wide column blocks (12000/16)
#define BPB    94               // uint4 mask batches per row (752 ushorts / 8)
#define JC     8                // J-chunks (partial buffers)
#define BPC    ((BPB + JC - 1) / JC)   // batches per chunk = 12
#define NELEM  (NN * DIM)       // 768000

typedef __attribute__((ext_vector_type(2))) float v2f;
typedef __attribute__((ext_vector_type(8))) float v8f;
typedef unsigned long long u64;

// ---------------- setup kernels ----------------
__global__ void k_zero_bitmap(unsigned* __restrict__ bm) {
    int n = NN * WPRP;
    for (int i = blockIdx.x * blockDim.x + threadIdx.x; i < n; i += gridDim.x * blockDim.x)
        bm[i] = 0u;
}

__global__ void k_scatter_edges(const int* __restrict__ eidx, unsigned* __restrict__ bm) {
    int e = blockIdx.x * blockDim.x + threadIdx.x;
    if (e >= NEDGES) return;
    int r = eidx[e];
    int c = eidx[NEDGES + e];
    if ((unsigned)r < NN && (unsigned)c < NN) {
        atomicOr(&bm[r * WPRP + (c >> 5)], 1u << (c & 31));
        atomicOr(&bm[c * WPRP + (r >> 5)], 1u << (r & 31));
    }
}

__global__ void k_degrees(const unsigned* __restrict__ bm, float* __restrict__ invd) {
    int r = blockIdx.x * blockDim.x + threadIdx.x;
    if (r >= NN) return;
    int deg = 0;
    const unsigned* row = bm + (size_t)r * WPRP;
    for (int j = 0; j < WPRP; ++j) deg += __popc(row[j]);   // padding word is zero
    float d = (float)deg;
    if (d < 1.0f) d = 1.0f;               // clip(deg, 1.0)
    invd[r] = rsqrtf(d);
}

// Xs = diag(invd) * X0 (column pre-scaling for the SpMM); acc = X0
__global__ void k_init_x(const float* __restrict__ ue, const float* __restrict__ ie,
                         const float* __restrict__ invd,
                         float* __restrict__ xs, float* __restrict__ acc) {
    int i = blockIdx.x * blockDim.x + threadIdx.x;
    if (i >= NELEM) return;
    float v = (i < NUSERS * DIM) ? ue[i] : ie[i - NUSERS * DIM];
    xs[i] = v * invd[i >> 6];
    acc[i] = v;
}

// ---------------- WMMA bitmap-SpMM ----------------
// Y = Â X = diag(invd) · A · (diag(invd)·X).  One wave owns a 16-row block I
// and a chunk of 16-col blocks J.  A-fragments are pure bit decompression
// (a = (float)bit, no loads/branches); the row scale invd_i is applied at the
// partial store.  Masks stream in as uint4 (8 J-blocks per b128 load) but the
// per-J loop stays ROLLED so the accumulator fragments coalesce in place
// (v_wmma D==C, no live-range-split mov storms).
__global__ __launch_bounds__(32) void k_spmm(const unsigned* __restrict__ bm,
                                             const float* __restrict__ invd,
                                             const float* __restrict__ Xs,
                                             float* __restrict__ partial) {
    const int I    = blockIdx.x;          // row block 0..NB-1
    const int jc   = blockIdx.y;          // J chunk  0..JC-1
    const int lane = threadIdx.x;
    const int lo   = lane & 15;           // M index of A / N index of B,D
    const int hi   = lane >> 4;           // K half selector

    const int row = I * 16 + lo;
    const uint4* mrow = (const uint4*)((const char*)bm + (size_t)row * ROWB);

    v8f d[4];
    #pragma unroll
    for (int nt = 0; nt < 4; ++nt) d[nt] = (v8f)0.0f;

    int b0 = jc * BPC;
    int b1 = b0 + BPC; if (b1 > BPB) b1 = BPB;

    #pragma unroll 1
    for (int b = b0; b < b1; ++b) {
        uint4 m4 = mrow[b];                       // masks for J = 8b .. 8b+7
        __builtin_prefetch(&mrow[b + 1], 0, 1);   // global_prefetch of next batch
        if (!__any((m4.x | m4.y | m4.z | m4.w) != 0u)) continue;

        u64 qlo = (u64)m4.x | ((u64)m4.y << 32);  // J = 8b+0 .. 8b+3
        u64 qhi = (u64)m4.z | ((u64)m4.w << 32);  // J = 8b+4 .. 8b+7

        #pragma unroll 1
        for (int p = 0; p < 2; ++p) {
            u64 q = p ? qhi : qlo;
            if (!__any(q != 0ull)) continue;

            #pragma unroll 1
            for (int jj = 0; jj < 4; ++jj) {
                unsigned mask = (unsigned)(q >> (16 * jj)) & 0xFFFFu;
                if (!__any(mask != 0u)) continue;             // wave-uniform skip
                const int J = b * 8 + p * 4 + jj;

                #pragma unroll
                for (int s = 0; s < 4; ++s) {                 // 4-wide K strips
                    unsigned sm = (mask >> (4 * s)) & 0xFu;
                    if (!__any(sm != 0u)) continue;           // wave-uniform skip

                    const int k0 = J * 16 + 4 * s + 2 * hi;   // this lane's K cols
                    v2f a;                                    // branch-free bit decompress
                    a.x = (float)((mask >> (4 * s + 2 * hi    )) & 1u);
                    a.y = (float)((mask >> (4 * s + 2 * hi + 1)) & 1u);

                    const float* xr = Xs + (size_t)k0 * DIM;  // rows k0, k0+1 of Xs
                    #pragma unroll
                    for (int nt = 0; nt < 4; ++nt) {
                        v2f bv;
                        bv.x = xr[       nt * 16 + lo];
                        bv.y = xr[DIM +  nt * 16 + lo];
                        d[nt] = __builtin_amdgcn_wmma_f32_16x16x4_f32(
                            false, a, false, bv, (short)0, d[nt], false, false);
                    }
                }
            }
        }
    }

    // D fragment layout: VGPR v -> row v + 8*hi, col nt*16 + lo.
    // Apply the left diag(invd) row scale here.
    float* po = partial + (size_t)jc * NELEM;
    #pragma unroll
    for (int v = 0; v < 8; ++v) {
        int r = I * 16 + v + 8 * hi;
        float irv = invd[r];
        #pragma unroll
        for (int nt = 0; nt < 4; ++nt)
            po[(size_t)r * DIM + nt * 16 + lo] = irv * d[nt][v];
    }
}

// deterministic reduction over JC partials + layer accumulation;
// also emits the invd-pre-scaled operand for the next layer
__global__ void k_reduce(const float* __restrict__ partial, const float* __restrict__ invd,
                         float* __restrict__ xnext_scaled, float* __restrict__ acc,
                         float scale) {
    int i = blockIdx.x * blockDim.x + threadIdx.x;
    if (i >= NELEM) return;
    float s = 0.0f;
    #pragma unroll
    for (int jc = 0; jc < JC; ++jc) s += partial[(size_t)jc * NELEM + i];
    xnext_scaled[i] = s * invd[i >> 6];
    acc[i] = (acc[i] + s) * scale;        // scale = 1, 1, 0.25 across layers
}

// ---------------- host-side launch ----------------
extern "C" void kernel_launch(void* const* d_in, const int* in_sizes, int n_in,
                              void* d_out, int out_size, void* d_ws, size_t ws_size,
                              hipStream_t stream) {
    const int*   eidx = (const int*)d_in[0];    // (2, NUM_EDGES) int32
    const float* ue   = (const float*)d_in[1];  // (8000, 64) f32
    const float* ie   = (const float*)d_in[2];  // (4000, 64) f32
    float* out = (float*)d_out;                 // (12000, 64) f32 accumulator/output

    char* ws = (char*)d_ws;
    size_t off = 0;
    unsigned* bitmap = (unsigned*)(ws + off); off += (size_t)NN * WPRP * 4;  // 18.05 MB
    float*    invd   = (float*)   (ws + off); off += (size_t)NN * 4;         // 48 KB
    float*    xa     = (float*)   (ws + off); off += (size_t)NELEM * 4;      // 3 MB
    float*    xb     = (float*)   (ws + off); off += (size_t)NELEM * 4;      // 3 MB
    float*    part   = (float*)   (ws + off); off += (size_t)JC * NELEM * 4; // 24.6 MB
    (void)ws_size; (void)in_sizes; (void)n_in; (void)out_size;

    // 1) adjacency bitmap + normalization
    k_zero_bitmap  <<<4096, 256, 0, stream>>>(bitmap);
    k_scatter_edges<<<(NEDGES + 255) / 256, 256, 0, stream>>>(eidx, bitmap);
    k_degrees      <<<(NN + 255) / 256, 256, 0, stream>>>(bitmap, invd);

    // 2) pre-scaled X0 and accumulator
    k_init_x<<<(NELEM + 255) / 256, 256, 0, stream>>>(ue, ie, invd, xa, out);

    // 3) three propagation layers: X_{l+1} = Â X_l ; acc += X_{l+1}
    float* xcur = xa;
    float* xnxt = xb;
    for (int l = 0; l < NLAYERS; ++l) {
        k_spmm<<<dim3(NB, JC), 32, 0, stream>>>(bitmap, invd, xcur, part);
        float scale = (l == NLAYERS - 1) ? 0.25f : 1.0f;   // final mean over 4 terms
        k_reduce<<<(NELEM + 255) / 256, 256, 0, stream>>>(part, invd, xnxt, out, scale);
        float* t = xcur; xcur = xnxt; xnxt = t;
    }
}